// EfficientRNN_52810917871801
// MI455X (gfx1250) — compile-verified
//
#include <hip/hip_runtime.h>
#include <math.h>

// Problem constants
#define T_DIM 4096
#define I_DIM 2048
#define H_DIM 2048
#define G3    (3 * H_DIM)   // 6144

// GEMM tiling
#define BM 128
#define BN 64
#define BK 32
// Row pad: stride 36 dwords (144B). 36*r mod 64 is distinct for r=0..15 ->
// conflict-free fragment reads; 144B is 16B-aligned for B128 async stores,
// and even-dword offsets keep ds_load_2addr_b64 legal.
#define LDK (BK + 4)

#define A_FLOATS (BM * LDK)          // 4608
#define B_FLOATS (3 * BN * LDK)      // 6912

typedef __attribute__((ext_vector_type(2))) float v2f;
typedef __attribute__((ext_vector_type(8))) float v8f;
typedef int v4i_vs __attribute__((vector_size(16)));   // builtin's pointee type

#if defined(__gfx1250__) && \
    __has_builtin(__builtin_amdgcn_global_load_async_to_lds_b128) && \
    __has_builtin(__builtin_amdgcn_s_wait_asynccnt)
#define USE_ASYNC_LDS 1
#else
#define USE_ASYNC_LDS 0
#endif

#define GPTR(p) ((v4i_vs*)(p))                                     // global src
#define LPTR(p) ((__attribute__((address_space(3))) v4i_vs*)(p))   // LDS dst

// ---------------------------------------------------------------------------
// Inner product phase: 8 k-steps x (3 gates x 2x2 tiles) of
// v_wmma_f32_16x16x4_f32 fed from LDS fragments.
// A frag: a[v] = A[lane&15][2*(lane>>4)+v]; B frag: b[v] = B[2*(lane>>4)+v][lane&15]
// ---------------------------------------------------------------------------
__device__ __forceinline__ void compute_tile(const float* __restrict__ cA,
                                             const float* __restrict__ cB,
                                             v8f acc[3][2][2],
                                             int lane, int wm, int wn)
{
    const int mrow = lane & 15;
    const int koff = (lane >> 4) << 1;   // lanes 16-31 hold K+2,K+3
    #pragma unroll
    for (int kk = 0; kk < BK; kk += 4) {
        v2f af[2];
        #pragma unroll
        for (int mi = 0; mi < 2; ++mi) {
            const float* p = &cA[(wm * 32 + mi * 16 + mrow) * LDK + kk + koff];
            af[mi][0] = p[0]; af[mi][1] = p[1];
        }
        v2f bf[3][2];
        #pragma unroll
        for (int g = 0; g < 3; ++g)
            #pragma unroll
            for (int ni = 0; ni < 2; ++ni) {
                const float* p = &cB[(g * BN + wn * 32 + ni * 16 + mrow) * LDK + kk + koff];
                bf[g][ni][0] = p[0]; bf[g][ni][1] = p[1];
            }
        #pragma unroll
        for (int g = 0; g < 3; ++g)
            #pragma unroll
            for (int mi = 0; mi < 2; ++mi)
                #pragma unroll
                for (int ni = 0; ni < 2; ++ni)
                    acc[g][mi][ni] = __builtin_amdgcn_wmma_f32_16x16x4_f32(
                        false, af[mi], false, bf[g][ni],
                        (short)0, acc[g][mi][ni], false, false);
    }
}

#if USE_ASYNC_LDS
// Stage one (kb) tile into LDS with GLOBAL_LOAD_ASYNC_TO_LDS_B128 (ASYNCcnt,
// no VGPR staging). 2560 16B chunks / 256 threads = 10 per thread.
__device__ __forceinline__ void stage_async(const float* __restrict__ x,
                                            const float* __restrict__ W_ih,
                                            size_t wbase, int t0, int h0, int kb,
                                            float* __restrict__ dA,
                                            float* __restrict__ dB, int tid)
{
    #pragma unroll
    for (int i = 0; i < 4; ++i) {                    // A: 1024 chunks
        int v  = tid + 256 * i;
        int r  = v >> 3;
        int c4 = (v & 7) << 2;
        __builtin_amdgcn_global_load_async_to_lds_b128(
            GPTR(x + (size_t)(t0 + r) * I_DIM + kb + c4),
            LPTR(dA + r * LDK + c4), 0, 0);
    }
    #pragma unroll
    for (int i = 0; i < 6; ++i) {                    // B: 1536 chunks
        int u   = tid + 256 * i;
        int g   = u >> 9;
        int rem = u & 511;
        int n   = rem >> 3;
        int c4  = (rem & 7) << 2;
        __builtin_amdgcn_global_load_async_to_lds_b128(
            GPTR(W_ih + wbase + (size_t)(g * H_DIM + h0 + n) * I_DIM + kb + c4),
            LPTR(dB + (g * BN + n) * LDK + c4), 0, 0);
    }
}
#endif

// ---------------------------------------------------------------------------
// Kernel 1: fused GEMM + GRU-gate epilogue with double-buffered async LDS.
// ---------------------------------------------------------------------------
__global__ __launch_bounds__(256)
void gru_gates_wmma(const float* __restrict__ x,
                    const float* __restrict__ W_ih,
                    const float* __restrict__ b_ih,
                    const float* __restrict__ b_hh,
                    float* __restrict__ H_all)
{
#if USE_ASYNC_LDS
    __shared__ float sA[2][A_FLOATS];
    __shared__ float sB[2][B_FLOATS];
#else
    __shared__ float sA[1][A_FLOATS];
    __shared__ float sB[1][B_FLOATS];
#endif

    const int t0 = blockIdx.x * BM;
    const int h0 = blockIdx.y * BN;
    const int k  = blockIdx.z;

    const int tid  = threadIdx.x;
    const int lane = tid & 31;
    const int wave = tid >> 5;
    const int wm   = wave & 3;    // 4 waves along M (32 rows each)
    const int wn   = wave >> 2;   // 2 waves along N (32 cols each)

    const v8f vzero = {0.f, 0.f, 0.f, 0.f, 0.f, 0.f, 0.f, 0.f};
    v8f acc[3][2][2];
    #pragma unroll
    for (int g = 0; g < 3; ++g)
        #pragma unroll
        for (int mi = 0; mi < 2; ++mi)
            #pragma unroll
            for (int ni = 0; ni < 2; ++ni)
                acc[g][mi][ni] = vzero;

    const size_t wbase = (size_t)k * G3 * I_DIM;

#if USE_ASYNC_LDS
    const int NIT = I_DIM / BK;
    stage_async(x, W_ih, wbase, t0, h0, 0, sA[0], sB[0], tid);
    for (int it = 0; it < NIT; ++it) {
        const int cur = it & 1;
        __builtin_amdgcn_s_wait_asynccnt(0);  // my chunks of buf[cur] are in LDS
        __syncthreads();                      // everyone's chunks landed; prev reads done
        if (it + 1 < NIT)                     // prefetch next tile under the WMMAs
            stage_async(x, W_ih, wbase, t0, h0, (it + 1) * BK,
                        sA[cur ^ 1], sB[cur ^ 1], tid);
        compute_tile(sA[cur], sB[cur], acc, lane, wm, wn);
    }
#else
    for (int kb = 0; kb < I_DIM; kb += BK) {
        #pragma unroll
        for (int i = 0; i < 4; ++i) {
            int v  = tid + 256 * i;
            int r  = v >> 3;
            int c4 = (v & 7) << 2;
            float4 val = *(const float4*)(x + (size_t)(t0 + r) * I_DIM + kb + c4);
            float* dst = &sA[0][r * LDK + c4];
            dst[0] = val.x; dst[1] = val.y; dst[2] = val.z; dst[3] = val.w;
        }
        #pragma unroll
        for (int i = 0; i < 6; ++i) {
            int v   = tid + 256 * i;
            int g   = v >> 9;
            int rem = v & 511;
            int n   = rem >> 3;
            int c4  = (rem & 7) << 2;
            float4 val = *(const float4*)(W_ih + wbase +
                          (size_t)(g * H_DIM + h0 + n) * I_DIM + kb + c4);
            float* dst = &sB[0][(g * BN + n) * LDK + c4];
            dst[0] = val.x; dst[1] = val.y; dst[2] = val.z; dst[3] = val.w;
        }
        __syncthreads();
        compute_tile(sA[0], sB[0], acc, lane, wm, wn);
        __syncthreads();
    }
#endif

    // ---- fused epilogue: r = sig(gr + bhr); z = sig(gz + bhz);
    //      n = tanh(gn + r*bhn); H_all = (1-z)*n
    const size_t bofs = (size_t)k * G3;
    #pragma unroll
    for (int mi = 0; mi < 2; ++mi) {
        #pragma unroll
        for (int ni = 0; ni < 2; ++ni) {
            const int h   = h0 + wn * 32 + ni * 16 + (lane & 15);
            const float bir = b_ih[bofs + h];
            const float biz = b_ih[bofs + H_DIM + h];
            const float bin = b_ih[bofs + 2 * H_DIM + h];
            const float bhr = b_hh[bofs + h];
            const float bhz = b_hh[bofs + H_DIM + h];
            const float bhn = b_hh[bofs + 2 * H_DIM + h];
            #pragma unroll
            for (int j = 0; j < 8; ++j) {
                const int t = t0 + wm * 32 + mi * 16 + ((lane >> 4) << 3) + j;
                const float r  = 1.f / (1.f + __expf(-(acc[0][mi][ni][j] + bir + bhr)));
                const float z  = 1.f / (1.f + __expf(-(acc[1][mi][ni][j] + biz + bhz)));
                const float nn = tanhf(acc[2][mi][ni][j] + bin + r * bhn);
                H_all[((size_t)t * 3 + k) * H_DIM + h] = (1.f - z) * nn;
            }
        }
    }
}

// ---------------------------------------------------------------------------
// Kernel 2: out[row][j] = dot(M[row], W[j]) + bias[j], j=0..2. One wave/row.
// ---------------------------------------------------------------------------
__global__ __launch_bounds__(256)
void dot3_kernel(const float* __restrict__ M, int rstride,
                 const float* __restrict__ W, int wstride,
                 const float* __restrict__ bias,
                 float* __restrict__ out, int nrows, int len)
{
    const int lane = threadIdx.x & 31;
    const int row  = blockIdx.x * 8 + (threadIdx.x >> 5);
    if (row >= nrows) return;
    const float* m = M + (size_t)row * rstride;
    float s0 = 0.f, s1 = 0.f, s2 = 0.f;
    for (int h = lane; h < len; h += 32) {
        const float v = m[h];
        s0 = fmaf(v, W[h], s0);
        s1 = fmaf(v, W[wstride + h], s1);
        s2 = fmaf(v, W[2 * wstride + h], s2);
    }
    #pragma unroll
    for (int off = 16; off; off >>= 1) {
        s0 += __shfl_xor(s0, off, 32);
        s1 += __shfl_xor(s1, off, 32);
        s2 += __shfl_xor(s2, off, 32);
    }
    if (lane == 0) {
        const float b0 = bias ? bias[0] : 0.f;
        const float b1 = bias ? bias[1] : 0.f;
        const float b2 = bias ? bias[2] : 0.f;
        out[(size_t)row * 3 + 0] = s0 + b0;
        out[(size_t)row * 3 + 1] = s1 + b1;
        out[(size_t)row * 3 + 2] = s2 + b2;
    }
}

// ---------------------------------------------------------------------------
// Kernel 3: 3-state automaton scan (parallel transition table + serial walk).
// ---------------------------------------------------------------------------
__global__ __launch_bounds__(256)
void scan_kernel(const float* __restrict__ L,     // [T][3 prev][3 j]
                 const float* __restrict__ selx,  // [T][3]
                 int* __restrict__ k_idx)
{
    __shared__ unsigned int nxt[T_DIM];   // 16 KB
    for (int t = 1 + threadIdx.x; t < T_DIM; t += 256) {
        unsigned int packed = 0;
        const float s0 = selx[t * 3 + 0];
        const float s1 = selx[t * 3 + 1];
        const float s2 = selx[t * 3 + 2];
        #pragma unroll
        for (int prev = 0; prev < 3; ++prev) {
            const float* Lp = L + ((size_t)(t - 1) * 3 + prev) * 3;
            const float l0 = Lp[0] + s0, l1 = Lp[1] + s1, l2 = Lp[2] + s2;
            int best = 0; float bv = l0;               // first-max like jnp.argmax
            if (l1 > bv) { best = 1; bv = l1; }
            if (l2 > bv) { best = 2; }
            packed |= (unsigned)best << (2 * prev);
        }
        nxt[t] = packed;
    }
    __syncthreads();
    if (threadIdx.x == 0) {
        int k = 0;
        k_idx[0] = 0;
        for (int t = 1; t < T_DIM; ++t) {
            k = (int)((nxt[t] >> (2 * k)) & 3u);
            k_idx[t] = k;
        }
    }
}

// ---------------------------------------------------------------------------
// Kernel 4: outputs[t] = H_all[t, k_idx[t]]; h_last = outputs[T-1].
// ---------------------------------------------------------------------------
__global__ __launch_bounds__(256)
void gather_kernel(const float* __restrict__ H_all,
                   const int* __restrict__ k_idx,
                   float* __restrict__ out)
{
    const int t = blockIdx.x;
    const int k = k_idx[t];
    const float4* src = (const float4*)(H_all + ((size_t)t * 3 + k) * H_DIM);
    float4* dst = (float4*)(out + (size_t)t * H_DIM);
    #pragma unroll
    for (int i = 0; i < 2; ++i) {
        const int idx = threadIdx.x + 256 * i;
        const float4 v = src[idx];
        dst[idx] = v;
        if (t == T_DIM - 1)
            ((float4*)(out + (size_t)T_DIM * H_DIM))[idx] = v;
    }
}

// ---------------------------------------------------------------------------
extern "C" void kernel_launch(void* const* d_in, const int* in_sizes, int n_in,
                              void* d_out, int out_size, void* d_ws, size_t ws_size,
                              hipStream_t stream)
{
    const float* x     = (const float*)d_in[0];   // (1,T,I)
    const float* W_ih  = (const float*)d_in[1];   // (3,3H,I)
    // d_in[2] = W_hh : present in inputs but unused by the reference math
    const float* b_ih  = (const float*)d_in[3];   // (3,3H)
    const float* b_hh  = (const float*)d_in[4];   // (3,3H)
    const float* W_sel = (const float*)d_in[5];   // (3,H+I)
    const float* b_sel = (const float*)d_in[6];   // (3,)
    float* out = (float*)d_out;                   // T*H outputs ++ H h_last

    // workspace layout (floats): H_all[T][3][H] | L[T][3][3] | selx[T][3] | k_idx[T]
    float* H_all = (float*)d_ws;
    float* L     = H_all + (size_t)T_DIM * 3 * H_DIM;
    float* selx  = L + (size_t)T_DIM * 9;
    int*   k_idx = (int*)(selx + (size_t)T_DIM * 3);

    dim3 g1(T_DIM / BM, H_DIM / BN, 3);
    gru_gates_wmma<<<g1, 256, 0, stream>>>(x, W_ih, b_ih, b_hh, H_all);

    const int WS = H_DIM + I_DIM;  // W_sel row stride
    // L[t][prev][j] = Wh[j] . H_all[t][prev]
    dot3_kernel<<<(T_DIM * 3) / 8, 256, 0, stream>>>(
        H_all, H_DIM, W_sel, WS, nullptr, L, T_DIM * 3, H_DIM);
    // sel_x[t][j] = Wx[j] . x[t] + b_sel[j]
    dot3_kernel<<<T_DIM / 8, 256, 0, stream>>>(
        x, I_DIM, W_sel + H_DIM, WS, b_sel, selx, T_DIM, I_DIM);

    scan_kernel<<<1, 256, 0, stream>>>(L, selx, k_idx);
    gather_kernel<<<T_DIM, 256, 0, stream>>>(H_all, k_idx, out);
}